// ConvTransposeNd_12781822673277
// MI455X (gfx1250) — compile-verified
//
#include <hip/hip_runtime.h>

typedef float v2f __attribute__((ext_vector_type(2)));
typedef float v8f __attribute__((ext_vector_type(8)));

#define CIN_   32
#define COUT_  32
#define SD     16            // input spatial extent per dim
#define OD     18            // output spatial extent per dim
#define TAPS   81            // 3^4
#define TAPS_PAD 84          // padded so 84 % 4 == 0 (pad taps have zero weight)
#define TCHUNK 21            // 84 / 4 K-chunks per cin
#define KCHUNKS (CIN_ * TCHUNK)          // 672 chunks of K=4
#define XCH   65536          // 16^4 floats per (b, cin) plane
#define OCH   104976         // 18^4 floats per (b, cout) plane
#define NPIX  (2 * OCH)      // 209952 flattened output pixels (incl. batch)
#define NBLOCKS (NPIX / 32)  // 6561 waves, 32 pixels each (exact)
#define WPELEMS (KCHUNKS * 2 * 32 * 2)   // 86016 floats of prepacked weights

// ---------------------------------------------------------------------------
// One-shot weight prepack into the exact WMMA 16x4 f32 A-operand lane layout:
//   float offset = kc*128 + half*64 + lane*2 + c2
//   lane<16 holds K-slots {0,1}, lane>=16 holds {2,3}; M row = half*16 + lane%16
// kc = cin*21 + tc ; tap = tc*4 + slot ; taps >= 81 are zero padding.
// ---------------------------------------------------------------------------
__global__ __launch_bounds__(256)
void prepack_w(const float* __restrict__ w, float* __restrict__ wp) {
    int idx = blockIdx.x * 256 + threadIdx.x;
    if (idx >= WPELEMS) return;
    int c2   = idx & 1;
    int lane = (idx >> 1) & 31;
    int half = (idx >> 6) & 1;
    int kc   = idx >> 7;
    int cin  = kc / TCHUNK;
    int tc   = kc - cin * TCHUNK;
    int slot = ((lane >> 4) << 1) | c2;      // K-slot 0..3
    int tap  = tc * 4 + slot;
    int cout = half * 16 + (lane & 15);
    float v = 0.0f;
    if (tap < TAPS) v = w[(cin * COUT_ + cout) * TAPS + tap];
    wp[idx] = v;
}

__device__ __forceinline__
void decomp_pixel(int P, int& b, int& p0, int& p1, int& p2, int& p3) {
    b = P / OCH;
    int r = P - b * OCH;
    p0 = r / (OD * OD * OD); r -= p0 * (OD * OD * OD);
    p1 = r / (OD * OD);      r -= p1 * (OD * OD);
    p2 = r / OD;
    p3 = r - p2 * OD;
}

// ---------------------------------------------------------------------------
// Main kernel: 1 wave/block, wave owns a 32(cout) x 32(pixel) tile:
// 4 f32 accumulators, A operands shared across the two N-tiles.
// K loop: 21 tap-chunks x 32 cin, four v_wmma_f32_16x16x4_f32 per step.
// ---------------------------------------------------------------------------
__global__ __launch_bounds__(32)
void convt4d_wmma(const float* __restrict__ x, const float* __restrict__ wp,
                  const float* __restrict__ bias, float* __restrict__ out) {
    __shared__ int tapTab[TAPS_PAD * 32];       // [tap][pixel-column 0..31]

    const int lane  = threadIdx.x & 31;
    const int n     = lane & 15;                // N column within a 16-wide tile
    const int Pbase = blockIdx.x * 32;

    // ---- build per-tap gather-offset table; each lane fills its own column
    {
        int b, p0, p1, p2, p3;
        decomp_pixel(Pbase + lane, b, p0, p1, p2, p3);
        for (int tap = 0; tap < TAPS_PAD; ++tap) {
            int e = -1;
            if (tap < TAPS) {
                int t0 = tap / 27;
                int t1 = (tap / 9) % 3;
                int t2 = (tap / 3) % 3;
                int t3 = tap % 3;
                int q0 = p0 - t0, q1 = p1 - t1, q2 = p2 - t2, q3 = p3 - t3;
                if ((unsigned)q0 < SD && (unsigned)q1 < SD &&
                    (unsigned)q2 < SD && (unsigned)q3 < SD)
                    e = ((q0 * SD + q1) * SD + q2) * SD + q3;
            }
            tapTab[tap * 32 + lane] = e;
        }
    }
    __syncthreads();

    // ---- per-lane pixel state for the two N-tiles this wave computes
    int bA, pA0, pA1, pA2, pA3;
    int bB, pB0, pB1, pB2, pB3;
    decomp_pixel(Pbase + n,      bA, pA0, pA1, pA2, pA3);
    decomp_pixel(Pbase + 16 + n, bB, pB0, pB1, pB2, pB3);

    const float* xbA = x + (size_t)bA * CIN_ * XCH;
    const float* xbB = x + (size_t)bB * CIN_ * XCH;

    v8f acc00 = {};  // couts  0..15, pixels base+0..15
    v8f acc10 = {};  // couts 16..31, pixels base+0..15
    v8f acc01 = {};  // couts  0..15, pixels base+16..31
    v8f acc11 = {};  // couts 16..31, pixels base+16..31

    const int cslot = (lane >> 4) << 1;          // K-slot base: 0 or 2

    for (int tc = 0; tc < TCHUNK; ++tc) {
        int tap0 = tc * 4 + cslot;
        int iqA0 = tapTab[ tap0      * 32 + n];
        int iqA1 = tapTab[(tap0 + 1) * 32 + n];
        int iqB0 = tapTab[ tap0      * 32 + 16 + n];
        int iqB1 = tapTab[(tap0 + 1) * 32 + 16 + n];
        unsigned sA0 = (iqA0 < 0) ? 0u : (unsigned)iqA0;
        unsigned sA1 = (iqA1 < 0) ? 0u : (unsigned)iqA1;
        unsigned sB0 = (iqB0 < 0) ? 0u : (unsigned)iqB0;
        unsigned sB1 = (iqB1 < 0) ? 0u : (unsigned)iqB1;

        const float* xa  = xbA;
        const float* xb2 = xbB;
        const float* wpc = wp + tc * 128 + lane * 2;   // kc = cin*21 + tc

        for (int cin = 0; cin < CIN_; ++cin) {
            float vA0 = xa[sA0];
            float vA1 = xa[sA1];
            float vB0 = xb2[sB0];
            float vB1 = xb2[sB1];
            v2f bbA = { (iqA0 < 0) ? 0.0f : vA0, (iqA1 < 0) ? 0.0f : vA1 };
            v2f bbB = { (iqB0 < 0) ? 0.0f : vB0, (iqB1 < 0) ? 0.0f : vB1 };

            v2f a0 = *(const v2f*)(wpc);               // couts 0..15
            v2f a1 = *(const v2f*)(wpc + 64);          // couts 16..31

            acc00 = __builtin_amdgcn_wmma_f32_16x16x4_f32(
                        false, a0, false, bbA, (short)0, acc00, false, false);
            acc10 = __builtin_amdgcn_wmma_f32_16x16x4_f32(
                        false, a1, false, bbA, (short)0, acc10, false, false);
            acc01 = __builtin_amdgcn_wmma_f32_16x16x4_f32(
                        false, a0, false, bbB, (short)0, acc01, false, false);
            acc11 = __builtin_amdgcn_wmma_f32_16x16x4_f32(
                        false, a1, false, bbB, (short)0, acc11, false, false);

            xa  += XCH;
            xb2 += XCH;
            wpc += TCHUNK * 128;                       // next cin's chunk block
        }
    }

    // ---- store D per 16x16 f32 C/D layout: VGPR j -> M = j + 8*(lane>=16)
    int spA = ((pA0 * OD + pA1) * OD + pA2) * OD + pA3;
    int spB = ((pB0 * OD + pB1) * OD + pB2) * OD + pB3;
    float* oA = out + (size_t)(bA * COUT_) * OCH + spA;
    float* oB = out + (size_t)(bB * COUT_) * OCH + spB;
    int mbase = (lane >> 4) << 3;                      // 0 or 8
#pragma unroll
    for (int j = 0; j < 8; ++j) {
        int c0 = mbase + j;
        float bi0 = bias[c0];
        float bi1 = bias[c0 + 16];
        oA[(size_t)c0 * OCH]        = acc00[j] + bi0;
        oA[(size_t)(c0 + 16) * OCH] = acc10[j] + bi1;
        oB[(size_t)c0 * OCH]        = acc01[j] + bi0;
        oB[(size_t)(c0 + 16) * OCH] = acc11[j] + bi1;
    }
}

extern "C" void kernel_launch(void* const* d_in, const int* in_sizes, int n_in,
                              void* d_out, int out_size, void* d_ws, size_t ws_size,
                              hipStream_t stream) {
    const float* x    = (const float*)d_in[0];
    const float* w    = (const float*)d_in[1];
    const float* bias = (const float*)d_in[2];
    float* out = (float*)d_out;
    float* wp  = (float*)d_ws;                         // 86016 floats = 344 KB

    prepack_w<<<(WPELEMS + 255) / 256, 256, 0, stream>>>(w, wp);
    convt4d_wmma<<<NBLOCKS, 32, 0, stream>>>(x, wp, bias, out);
}